// AdaFocal_5970004542007
// MI455X (gfx1250) — compile-verified
//
#include <hip/hip_runtime.h>

typedef __attribute__((ext_vector_type(2))) float v2f;
typedef __attribute__((ext_vector_type(8))) float v8f;

#define NBINS   15
#define LOG2E   1.4426950408889634f
#define LN2     0.6931471805599453f
#define F32EPS  1.1920928955078125e-07f   // 2^-23, matches np.finfo(float32).eps

// One AdaFocal element:
//   ce  = softplus(x) - t*x                      (BCE-with-logits simplification)
//   pt  = t ? sigmoid(x) : 1 - sigmoid(x)
//   g   = gammas[clip(int(pt*15), 0, 14)]
//   out = ce * (1 - sign(g)*pt + eps)^|g|
__device__ __forceinline__ float adafocal_elem(float x, int t, const float* __restrict__ sgam) {
    float ax = fabsf(x);
    float e2 = __builtin_amdgcn_exp2f(-ax * LOG2E);            // exp(-|x|)   (v_exp_f32)
    float lp = __builtin_amdgcn_logf(1.0f + e2) * LN2;         // log1p(e2)   (v_log_f32)
    float sp = lp + fmaxf(x, 0.0f);                            // softplus(x)
    float r  = __builtin_amdgcn_rcpf(1.0f + e2);               // 1/(1+e2)    (v_rcp_f32)

    float tf = (float)t;
    float ce = sp - tf * x;

    // sigmoid branches share e2: x>=0 -> p=r ; x<0 -> p=e2*r.  pt = t ? p : 1-p
    bool  cond = (t != 0) == (x >= 0.0f);
    float pt   = cond ? r : e2 * r;

    int bin = (int)(pt * (float)NBINS);                        // trunc == floor (pt >= 0)
    bin = bin < 0 ? 0 : (bin > NBINS - 1 ? NBINS - 1 : bin);
    float g = sgam[bin];

    float sgn  = (g > 0.0f) ? 1.0f : ((g < 0.0f) ? -1.0f : 0.0f);
    float mag  = fabsf(g);
    float base = 1.0f - sgn * pt + F32EPS;                     // always > 0
    // pow(base, mag); fast path mag==1 (the provided gammas are all 1.0)
    float w = (mag == 1.0f)
                ? base
                : __builtin_amdgcn_exp2f(mag * __builtin_amdgcn_logf(base));
    return ce * w;
}

__global__ void __launch_bounds__(256) adafocal_partial_kernel(
    const float4* __restrict__ x4, const int4* __restrict__ t4,
    const float* __restrict__ bin_gammas,
    float* __restrict__ partials, int nvec, int ntail_base, int ntail) {
    __shared__ float sgam[NBINS];
    __shared__ float swsum[8];

    if (threadIdx.x < NBINS) sgam[threadIdx.x] = bin_gammas[threadIdx.x];
    __syncthreads();

    const int stride = gridDim.x * blockDim.x;
    float acc = 0.0f;

    for (int i = blockIdx.x * blockDim.x + threadIdx.x; i < nvec; i += stride) {
        // Speculative prefetch a few strides ahead (gfx1250 global_prefetch_b8);
        // OOB speculative prefetches are silently dropped by hardware.
        __builtin_prefetch(&x4[i + 4 * stride], 0, 0);
        __builtin_prefetch(&t4[i + 4 * stride], 0, 0);

        float4 x = x4[i];
        int4   t = t4[i];
        acc += adafocal_elem(x.x, t.x, sgam);
        acc += adafocal_elem(x.y, t.y, sgam);
        acc += adafocal_elem(x.z, t.z, sgam);
        acc += adafocal_elem(x.w, t.w, sgam);
    }

    // Scalar tail (n not divisible by 4) — handled by block 0, thread 0 only.
    if (blockIdx.x == 0 && threadIdx.x == 0 && ntail > 0) {
        const float* xs = (const float*)x4;
        const int*   ts = (const int*)t4;
        for (int i = 0; i < ntail; ++i)
            acc += adafocal_elem(xs[ntail_base + i], ts[ntail_base + i], sgam);
    }

    // wave32 butterfly reduction
    for (int m = 16; m >= 1; m >>= 1)
        acc += __shfl_xor(acc, m, 32);

    const int wave = threadIdx.x >> 5;
    if ((threadIdx.x & 31) == 0) swsum[wave] = acc;
    __syncthreads();

    if (threadIdx.x == 0) {
        float s = 0.0f;
        #pragma unroll
        for (int w = 0; w < 8; ++w) s += swsum[w];
        partials[blockIdx.x] = s;
    }
}

// Single-wave final reduction. Lane sums are folded on the matrix pipe with
// V_WMMA_F32_16X16X4_F32: A = ones(16x4) so D[m][n] = sum_k B[k][n]; each
// lane's value occupies exactly one B cell, so the 16 column sums partition
// all 32 lanes regardless of B striping details. EXEC is all-1s (32 threads).
__global__ void __launch_bounds__(32) adafocal_final_kernel(
    const float* __restrict__ partials, int n, float* __restrict__ out) {
    float s = 0.0f;
    for (int i = threadIdx.x; i < n; i += 32)
        s += partials[i];

    v2f a; a[0] = 1.0f; a[1] = 1.0f;        // all-ones A (both K slots this lane owns)
    v2f b; b[0] = s;    b[1] = 0.0f;        // one live cell per lane
    v8f c = {};
    c = __builtin_amdgcn_wmma_f32_16x16x4_f32(
            /*neg_a=*/false, a, /*neg_b=*/false, b,
            /*c_mod=*/(short)0, c, /*reuse_a=*/false, /*reuse_b=*/false);

    // c[0]: lanes 0..15 hold D[0][n] (column sums); fold those 16 lanes.
    float v = c[0];
    for (int m = 8; m >= 1; m >>= 1)
        v += __shfl_xor(v, m, 16);

    if (threadIdx.x == 0) out[0] = v;
}

extern "C" void kernel_launch(void* const* d_in, const int* in_sizes, int n_in,
                              void* d_out, int out_size, void* d_ws, size_t ws_size,
                              hipStream_t stream) {
    const float4* x4  = (const float4*)d_in[0];   // inputs,  f32, 16384*2048
    const int4*   t4  = (const int4*)d_in[1];     // targets, i32, 16384*2048
    const float*  gam = (const float*)d_in[2];    // bin_gammas, f32, 15
    float* out      = (float*)d_out;              // scalar sum
    float* partials = (float*)d_ws;               // 2048 floats of scratch

    const int n     = in_sizes[0];
    const int nvec  = n >> 2;
    const int ntail = n & 3;

    const int THREADS = 256;
    const int BLOCKS  = 2048;

    adafocal_partial_kernel<<<BLOCKS, THREADS, 0, stream>>>(
        x4, t4, gam, partials, nvec, nvec << 2, ntail);
    adafocal_final_kernel<<<1, 32, 0, stream>>>(partials, BLOCKS, out);
}